// GCNN_15556371547005
// MI455X (gfx1250) — compile-verified
//
#include <hip/hip_runtime.h>

#define NNODES 40000
#define NEDGES 640000
#define FIN    128
#define HDIM   512
#define NGRAPH 64

typedef __attribute__((ext_vector_type(2))) float v2f;
typedef __attribute__((ext_vector_type(8))) float v8f;

// fp32 atomic add that is guaranteed to lower to GLOBAL_ATOMIC_ADD_F32
__device__ __forceinline__ void fadd_atomic(float* p, float v) {
    unsafeAtomicAdd(p, v);
}

// ---------------------------------------------------------------- zero
__global__ void zero_kernel(float4* __restrict__ p, int nvec4) {
    int i = blockIdx.x * blockDim.x + threadIdx.x;
    if (i < nvec4) p[i] = make_float4(0.f, 0.f, 0.f, 0.f);
}

// ---------------------------------------------------------------- scatter: agg[dst] += x[src] * ew
template <int C>
__global__ __launch_bounds__(256)
void scatter_kernel(const float* __restrict__ xin,
                    const int*   __restrict__ ei,
                    const float* __restrict__ ew,
                    float*       __restrict__ agg) {
    const int CV  = C / 4;
    int tid = blockIdx.x * blockDim.x + threadIdx.x;
    if (tid >= NEDGES * CV) return;
    int e = tid / CV;
    int c = (tid - e * CV) * 4;
    int s = ei[e];
    int d = ei[NEDGES + e];
    float w = ew[e];
    const float4 v = *reinterpret_cast<const float4*>(xin + (size_t)s * C + c);
    float* p = agg + (size_t)d * C + c;
    fadd_atomic(p + 0, v.x * w);
    fadd_atomic(p + 1, v.y * w);
    fadd_atomic(p + 2, v.z * w);
    fadd_atomic(p + 3, v.w * w);
}

// ---------------------------------------------------------------- fused dual-GEMM + bias + relu
// out[m,n] = relu( sum_k A1[m,k] W1[k,n] + sum_k A2[m,k] W2[k,n] + bias[n] ), n in [0,512)
// ldsA holds a 16-row A tile transposed: ldsA[k*16 + m]
__device__ __forceinline__
void wmma_section(const float* __restrict__ ldsA, const float* __restrict__ W,
                  int K, int nbase, int m, int hi, v8f acc[4]) {
    for (int k0 = 0; k0 < K; k0 += 4) {
        const int kk = k0 + 2 * hi;                    // half-wave K split per A layout
        v2f a;
        a.x = ldsA[kk * 16 + m];                       // A[m][kk]
        a.y = ldsA[kk * 16 + 16 + m];                  // A[m][kk+1]
        const float* wr0 = W + (size_t)kk * HDIM + nbase + m;
        const float* wr1 = wr0 + HDIM;
        // prefetch next k-step's B rows (global_prefetch_b8) to hide L2 latency
        __builtin_prefetch(wr0 + 4 * HDIM, 0, 1);
        __builtin_prefetch(wr1 + 4 * HDIM, 0, 1);
#pragma unroll
        for (int t = 0; t < 4; ++t) {
            v2f b;
            b.x = wr0[16 * t];                         // B[kk][n]
            b.y = wr1[16 * t];                         // B[kk+1][n]
            acc[t] = __builtin_amdgcn_wmma_f32_16x16x4_f32(
                false, a, false, b, (short)0, acc[t], false, false);
        }
    }
}

__global__ __launch_bounds__(256)
void gemm_fused_kernel(const float* __restrict__ A1, const float* __restrict__ W1, int K1,
                       const float* __restrict__ A2, const float* __restrict__ W2, int K2,
                       const float* __restrict__ bias, float* __restrict__ out, int do_relu) {
    __shared__ float lds[16 * 1024];                   // 64 KB: [K1][16] then [K2][16]
    const int mbase = blockIdx.x * 16;
    const int tid   = threadIdx.x;

    for (int idx = tid; idx < 16 * K1; idx += 256) {   // coalesced global read, transposed store
        int r = idx / K1, c = idx - r * K1;
        lds[c * 16 + r] = A1[(size_t)(mbase + r) * K1 + c];
    }
    float* lds2 = lds + 16 * K1;
    for (int idx = tid; idx < 16 * K2; idx += 256) {
        int r = idx / K2, c = idx - r * K2;
        lds2[c * 16 + r] = A2[(size_t)(mbase + r) * K2 + c];
    }
    __syncthreads();

    const int lane  = tid & 31;
    const int wave  = tid >> 5;
    const int nbase = wave * 64;                       // 8 waves cover 512 cols
    const int m     = lane & 15;
    const int hi    = lane >> 4;

    v8f acc[4] = {};
    wmma_section(lds,  W1, K1, nbase, m, hi, acc);
    wmma_section(lds2, W2, K2, nbase, m, hi, acc);

    const int n0 = nbase + m;
#pragma unroll
    for (int t = 0; t < 4; ++t) {
        const int n  = n0 + 16 * t;
        const float bv = bias[n];
#pragma unroll
        for (int r = 0; r < 8; ++r) {
            int row = mbase + r + 8 * hi;              // C/D layout: VGPR r -> M=r(+8)
            float v = acc[t][r] + bv;
            if (do_relu) v = fmaxf(v, 0.0f);
            out[(size_t)row * HDIM + n] = v;
        }
    }
}

// ---------------------------------------------------------------- global mean pool (sum + count)
__global__ __launch_bounds__(256)
void pool_kernel(const float* __restrict__ h, const int* __restrict__ batch,
                 float* __restrict__ sums, float* __restrict__ cnt) {
    const int CV = HDIM / 4;
    int tid = blockIdx.x * blockDim.x + threadIdx.x;
    if (tid >= NNODES * CV) return;
    int i = tid / CV;
    int c = (tid - i * CV) * 4;
    int g = batch[i];
    const float4 v = *reinterpret_cast<const float4*>(h + (size_t)i * HDIM + c);
    float* p = sums + (size_t)g * HDIM + c;
    fadd_atomic(p + 0, v.x);
    fadd_atomic(p + 1, v.y);
    fadd_atomic(p + 2, v.z);
    fadd_atomic(p + 3, v.w);
    if (c == 0) fadd_atomic(cnt + g, 1.0f);
}

// ---------------------------------------------------------------- MLP head (tiny, one block)
__global__ __launch_bounds__(256)
void head_kernel(const float* __restrict__ sums, const float* __restrict__ cnt,
                 const float* __restrict__ W1, const float* __restrict__ b1,
                 const float* __restrict__ W2, const float* __restrict__ b2,
                 const float* __restrict__ W3, const float* __restrict__ b3,
                 float* __restrict__ out) {
    __shared__ float o1[NGRAPH * 64];
    __shared__ float o2[NGRAPH * 16];
    const int tid = threadIdx.x;
    for (int o = tid; o < NGRAPH * 64; o += 256) {
        int g = o >> 6, j = o & 63;
        float inv = 1.0f / fmaxf(cnt[g], 1.0f);
        float s = b1[j];
        for (int c = 0; c < HDIM; ++c)
            s += (sums[(size_t)g * HDIM + c] * inv) * W1[c * 64 + j];
        o1[o] = fmaxf(s, 0.0f);
    }
    __syncthreads();
    for (int o = tid; o < NGRAPH * 16; o += 256) {
        int g = o >> 4, j = o & 15;
        float s = b2[j];
        for (int c = 0; c < 64; ++c)
            s += o1[g * 64 + c] * W2[c * 16 + j];
        o2[o] = fmaxf(s, 0.0f);
    }
    __syncthreads();
    if (tid < NGRAPH) {
        float s = b3[0];
        for (int c = 0; c < 16; ++c)
            s += o2[tid * 16 + c] * W3[c];
        out[tid] = s;
    }
}

// ---------------------------------------------------------------- launcher
extern "C" void kernel_launch(void* const* d_in, const int* in_sizes, int n_in,
                              void* d_out, int out_size, void* d_ws, size_t ws_size,
                              hipStream_t stream) {
    (void)in_sizes; (void)n_in; (void)out_size; (void)ws_size;

    const float* x   = (const float*)d_in[0];
    const int*   ei  = (const int*)  d_in[1];
    const float* ew  = (const float*)d_in[2];
    const int*   bat = (const int*)  d_in[3];
    const float* Wr1 = (const float*)d_in[4];
    const float* br1 = (const float*)d_in[5];
    const float* Wo1 = (const float*)d_in[6];
    const float* Wr2 = (const float*)d_in[7];
    const float* br2 = (const float*)d_in[8];
    const float* Wo2 = (const float*)d_in[9];
    const float* W1  = (const float*)d_in[10];
    const float* b1  = (const float*)d_in[11];
    const float* W2  = (const float*)d_in[12];
    const float* b2  = (const float*)d_in[13];
    const float* W3  = (const float*)d_in[14];
    const float* b3  = (const float*)d_in[15];

    float* ws   = (float*)d_ws;
    float* agg1 = ws;                                   // N*128
    float* agg2 = agg1 + (size_t)NNODES * FIN;          // N*512
    float* sums = agg2 + (size_t)NNODES * HDIM;         // 64*512
    float* cnt  = sums + (size_t)NGRAPH * HDIM;         // 64
    float* h1   = cnt  + NGRAPH;                        // N*512
    float* h2   = h1   + (size_t)NNODES * HDIM;         // N*512

    // zero agg1 | agg2 | sums | cnt (contiguous)
    const size_t nzero = (size_t)NNODES * FIN + (size_t)NNODES * HDIM
                       + (size_t)NGRAPH * HDIM + NGRAPH;
    const int nvec4 = (int)(nzero / 4);
    zero_kernel<<<(nvec4 + 255) / 256, 256, 0, stream>>>((float4*)ws, nvec4);

    // layer 1
    scatter_kernel<FIN><<<(NEDGES * (FIN / 4) + 255) / 256, 256, 0, stream>>>(x, ei, ew, agg1);
    gemm_fused_kernel<<<NNODES / 16, 256, 0, stream>>>(agg1, Wr1, FIN, x, Wo1, FIN, br1, h1, 1);

    // layer 2
    scatter_kernel<HDIM><<<(NEDGES * (HDIM / 4) + 255) / 256, 256, 0, stream>>>(h1, ei, ew, agg2);
    gemm_fused_kernel<<<NNODES / 16, 256, 0, stream>>>(agg2, Wr2, HDIM, h1, Wo2, HDIM, br2, h2, 1);

    // pool + head
    pool_kernel<<<(NNODES * (HDIM / 4) + 255) / 256, 256, 0, stream>>>(h2, bat, sums, cnt);
    head_kernel<<<1, 256, 0, stream>>>(sums, cnt, W1, b1, W2, b2, W3, b3, (float*)d_out);
}